// ProposalGenerator_69836168233059
// MI455X (gfx1250) — compile-verified
//
#include <hip/hip_runtime.h>
#include <math.h>

#define PRE_NMS    6000
#define POST_NMS   1000
#define NMS_THR    0.7f
#define NBUCKET    4096
#define BSHIFT     20
#define CAND_CAP   8192   // sort size (power of two, >= PRE_NMS + slack)
#define NMS_THREADS 1024
#define SLOTS      6      // ceil(6000/1024)
#define NW         188    // ceil(6000/32) alive-bitmask words
#define NW_PAD     192

#if defined(__has_builtin)
#  define HAVE_ASYNC_LDS  __has_builtin(__builtin_amdgcn_global_load_async_to_lds_b64)
#  define HAVE_ASYNC_WAIT __has_builtin(__builtin_amdgcn_s_wait_asynccnt)
#else
#  define HAVE_ASYNC_LDS  0
#  define HAVE_ASYNC_WAIT 0
#endif

#if HAVE_ASYNC_LDS
typedef int v2i_t __attribute__((vector_size(2 * sizeof(int))));
typedef __attribute__((address_space(1))) v2i_t* g_v2i_p;   // global source
typedef __attribute__((address_space(3))) v2i_t* l_v2i_p;   // LDS destination
#endif

// ---------------- decode helpers (shared by all passes) ----------------

__device__ __forceinline__ void decode_box(const float4 a, const float4 d,
                                           const float W, const float H,
                                           float& x1, float& y1, float& x2, float& y2,
                                           bool& valid) {
    float aw = a.z - a.x;
    float ah = a.w - a.y;
    float ax = a.x + 0.5f * aw;
    float ay = a.y + 0.5f * ah;
    float px = ax + d.x * aw;
    float py = ay + d.y * ah;
    float pw = aw * expf(fminf(fmaxf(d.z, -10.f), 10.f));
    float ph = ah * expf(fminf(fmaxf(d.w, -10.f), 10.f));
    x1 = fminf(fmaxf(px - 0.5f * pw, 0.f), W - 1.f);
    y1 = fminf(fmaxf(py - 0.5f * ph, 0.f), H - 1.f);
    x2 = fminf(fmaxf(px + 0.5f * pw, 0.f), W - 1.f);
    y2 = fminf(fmaxf(py + 0.5f * ph, 0.f), H - 1.f);
    valid = ((x2 - x1) >= 16.0f) && ((y2 - y1) >= 16.0f);
}

// monotone key: larger float score -> larger unsigned key
__device__ __forceinline__ unsigned score_key(float lg, bool valid) {
    float fg = valid ? (1.0f / (1.0f + expf(-lg))) : -1.0f;
    unsigned u = __float_as_uint(fg);
    return (u & 0x80000000u) ? ~u : (u | 0x80000000u);
}

// ---------------- kernel 0: zero workspace counters ----------------

__global__ void k_zero(int* __restrict__ p, int n) {
    int i = blockIdx.x * blockDim.x + threadIdx.x;
    if (i < n) p[i] = 0;
}

// ---------------- kernel 1: decode + 4096-bucket key histogram ----------------

__global__ void k_hist(const float4* __restrict__ anch, const int* __restrict__ isz,
                       const float* __restrict__ lgt, const float4* __restrict__ dlt,
                       int N, int* __restrict__ hist) {
    __shared__ int lh[NBUCKET];
    const int b = blockIdx.y;
    for (int i = threadIdx.x; i < NBUCKET; i += blockDim.x) lh[i] = 0;
    __syncthreads();

    const float H = (float)isz[b * 2 + 0];
    const float W = (float)isz[b * 2 + 1];
    const float4* A = anch + (size_t)b * N;
    const float4* D = dlt + (size_t)b * N;
    const float*  L = lgt + (size_t)b * N;

    const int stride = gridDim.x * blockDim.x;
    for (int i = blockIdx.x * blockDim.x + threadIdx.x; i < N; i += stride) {
        __builtin_prefetch((const void*)(A + i + stride), 0, 3);
        __builtin_prefetch((const void*)(D + i + stride), 0, 3);
        float4 a = A[i], d = D[i];
        float x1, y1, x2, y2; bool v;
        decode_box(a, d, W, H, x1, y1, x2, y2, v);
        unsigned key = score_key(L[i], v);
        atomicAdd(&lh[key >> BSHIFT], 1);
    }
    __syncthreads();

    int* gh = hist + (size_t)b * NBUCKET;
    for (int i = threadIdx.x; i < NBUCKET; i += blockDim.x) {
        int c = lh[i];
        if (c) atomicAdd(&gh[i], c);
    }
}

// ---------------- kernel 2: per-image threshold bucket ----------------

__global__ void k_thresh(const int* __restrict__ hist, int* __restrict__ meta, int B) {
    int b = blockIdx.x * blockDim.x + threadIdx.x;
    if (b >= B) return;
    const int* h = hist + (size_t)b * NBUCKET;
    int cum = 0, T = 0;
    for (int bk = NBUCKET - 1; bk >= 0; --bk) {
        cum += h[bk];
        if (cum >= PRE_NMS) { T = bk; break; }
    }
    meta[b] = T;
}

// ---------------- kernel 3: compact candidates, pre-packed (key<<32 | ~idx) ----------------

__global__ void k_compact(const float4* __restrict__ anch, const int* __restrict__ isz,
                          const float* __restrict__ lgt, const float4* __restrict__ dlt,
                          int N, const int* __restrict__ meta, int* __restrict__ cnt,
                          unsigned long long* __restrict__ candPacked) {
    const int b = blockIdx.y;
    const float H = (float)isz[b * 2 + 0];
    const float W = (float)isz[b * 2 + 1];
    const unsigned T = (unsigned)meta[b];
    const float4* A = anch + (size_t)b * N;
    const float4* D = dlt + (size_t)b * N;
    const float*  L = lgt + (size_t)b * N;
    unsigned long long* CP = candPacked + (size_t)b * CAND_CAP;

    const int stride = gridDim.x * blockDim.x;
    for (int i = blockIdx.x * blockDim.x + threadIdx.x; i < N; i += stride) {
        __builtin_prefetch((const void*)(A + i + stride), 0, 3);
        __builtin_prefetch((const void*)(D + i + stride), 0, 3);
        float4 a = A[i], d = D[i];
        float x1, y1, x2, y2; bool v;
        decode_box(a, d, W, H, x1, y1, x2, y2, v);
        unsigned key = score_key(L[i], v);
        if ((key >> BSHIFT) >= T) {
            int p = atomicAdd(&cnt[b], 1);
            if (p < CAND_CAP)
                CP[p] = ((unsigned long long)key << 32) | (unsigned)(~(unsigned)i);
        }
    }
}

// ---------------- kernel 4: bitonic sort (exact top-k) + sequential NMS + output ----------------

__global__ void __launch_bounds__(NMS_THREADS)
k_sort_nms_out(const float4* __restrict__ anch, const int* __restrict__ isz,
               const float* __restrict__ lgt, const float4* __restrict__ dlt,
               int N, const int* __restrict__ cnt,
               const unsigned long long* __restrict__ candPacked,
               float* __restrict__ outP, float* __restrict__ outL,
               float* __restrict__ outD, float* __restrict__ outV) {
    __shared__ __align__(16) unsigned char smem[65536];
    unsigned long long* sortBuf = (unsigned long long*)smem;          // [0, 65536)
    unsigned* sortedIdx = (unsigned*)smem;                            // [0, 24000)  (post-sort)
    unsigned* aliveW    = (unsigned*)(smem + 24064);                  // 192 words
    int*      keepArr   = (int*)(smem + 24832);                       // 1000 ints
    float*    bc        = (float*)(smem + 28832);                     // 5 floats

    const int b   = blockIdx.x;
    const int tid = threadIdx.x;
    const float H = (float)isz[b * 2 + 0];
    const float W = (float)isz[b * 2 + 1];
    const float4* A = anch + (size_t)b * N;
    const float4* D = dlt + (size_t)b * N;
    const float*  L = lgt + (size_t)b * N;

    // ---- fetch packed candidates into LDS (async DMA path on gfx1250) ----
    int M = cnt[b]; if (M > CAND_CAP) M = CAND_CAP;
    const unsigned long long* CP = candPacked + (size_t)b * CAND_CAP;
#if HAVE_ASYNC_LDS
    for (int i = tid; i < M; i += NMS_THREADS) {
        __builtin_amdgcn_global_load_async_to_lds_b64(
            (g_v2i_p)(CP + i), (l_v2i_p)(sortBuf + i), 0, 0);
    }
    for (int i = M + tid; i < CAND_CAP; i += NMS_THREADS) sortBuf[i] = 0ull;
#  if HAVE_ASYNC_WAIT
    __builtin_amdgcn_s_wait_asynccnt(0);
#  else
    asm volatile("s_wait_asynccnt 0x0" ::: "memory");
#  endif
#else
    for (int i = tid; i < CAND_CAP; i += NMS_THREADS)
        sortBuf[i] = (i < M) ? CP[i] : 0ull;
#endif

    // ---- bitonic sort, descending; ~idx low word => smaller index first on ties ----
    for (unsigned k = 2; k <= CAND_CAP; k <<= 1) {
        for (unsigned j = k >> 1; j > 0; j >>= 1) {
            __syncthreads();
            for (unsigned t = tid; t < CAND_CAP; t += NMS_THREADS) {
                unsigned l = t ^ j;
                if (l > t) {
                    unsigned long long va = sortBuf[t], vb = sortBuf[l];
                    bool desc = ((t & k) == 0);
                    if (desc ? (va < vb) : (va > vb)) { sortBuf[t] = vb; sortBuf[l] = va; }
                }
            }
        }
    }
    __syncthreads();

    // ---- pull my 6 sorted entries into registers ----
    unsigned rkey[SLOTS], rorig[SLOTS];
    float rx1[SLOTS], ry1[SLOTS], rx2[SLOTS], ry2[SLOTS], rar[SLOTS];
#pragma unroll
    for (int s = 0; s < SLOTS; ++s) {
        int j = s * NMS_THREADS + tid;
        rkey[s] = 0u; rorig[s] = 0u;
        rx1[s] = 0.f; ry1[s] = 0.f; rx2[s] = 0.f; ry2[s] = 0.f; rar[s] = 0.f;
        if (j < PRE_NMS) {
            unsigned long long v = sortBuf[j];
            rkey[s]  = (unsigned)(v >> 32);
            rorig[s] = ~((unsigned)(v & 0xFFFFFFFFull));
        }
    }
    __syncthreads();

    // ---- repurpose LDS: sortedIdx table + alive bitmask ----
#pragma unroll
    for (int s = 0; s < SLOTS; ++s) {
        int j = s * NMS_THREADS + tid;
        if (j < PRE_NMS) sortedIdx[j] = rorig[s];
    }
    for (int w = tid; w < NW_PAD; w += NMS_THREADS) aliveW[w] = 0u;
    __syncthreads();

#pragma unroll
    for (int s = 0; s < SLOTS; ++s) {
        int j = s * NMS_THREADS + tid;
        if (j < PRE_NMS && rkey[s] >= 0x80000000u)      // score > -0.5 (a real sigmoid)
            atomicOr(&aliveW[j >> 5], 1u << (j & 31));
        if (j < PRE_NMS) {                               // decode my boxes into registers
            unsigned oi = rorig[s];
            float4 a = A[oi], d = D[oi];
            bool v;
            decode_box(a, d, W, H, rx1[s], ry1[s], rx2[s], ry2[s], v);
            rar[s] = (rx2[s] - rx1[s]) * (ry2[s] - ry1[s]);
        }
    }
    __syncthreads();

    // ---- sequential NMS, 2 barriers/step ----
    // first-alive index is monotone non-decreasing (bits only ever cleared), and the
    // scan is pure LDS reads -> every thread runs it redundantly with a private cursor,
    // so all threads agree on fi with no extra barrier / atomicMin.
    int cw = 0;
    for (int step = 0; step < POST_NMS; ++step) {
        while (cw < NW && aliveW[cw] == 0u) ++cw;
        const int fi = (cw < NW) ? cw * 32 + (__ffs((int)aliveW[cw]) - 1) : -1;

        if (fi < 0) {                                    // nothing alive: fill & stop
            for (int s2 = step + tid; s2 < POST_NMS; s2 += NMS_THREADS) keepArr[s2] = -1;
            break;                                       // uniform across the block
        }
        if (tid == 0) keepArr[step] = fi;
        if ((fi & (NMS_THREADS - 1)) == tid) {           // owner broadcasts box (reads only of alive)
            int s = fi >> 10;
            bc[0] = rx1[s]; bc[1] = ry1[s]; bc[2] = rx2[s]; bc[3] = ry2[s]; bc[4] = rar[s];
        }
        __syncthreads();

        // suppression phase (alive mutations happen only here)
        if ((fi & (NMS_THREADS - 1)) == tid)
            atomicAnd(&aliveW[fi >> 5], ~(1u << (fi & 31)));
        {
            const float sx1 = bc[0], sy1 = bc[1], sx2 = bc[2], sy2 = bc[3], sa = bc[4];
#pragma unroll
            for (int s = 0; s < SLOTS; ++s) {
                int j = s * NMS_THREADS + tid;
                if (j < PRE_NMS) {
                    float iw = fmaxf(0.f, fminf(rx2[s], sx2) - fmaxf(rx1[s], sx1));
                    float ih = fmaxf(0.f, fminf(ry2[s], sy2) - fmaxf(ry1[s], sy1));
                    float inter = iw * ih;
                    float uni = rar[s] + sa - inter;
                    // iou > thr  <=>  inter > thr*uni   (areas >= 0, no division)
                    if (inter > NMS_THR * uni) atomicAnd(&aliveW[j >> 5], ~(1u << (j & 31)));
                }
            }
        }
        __syncthreads();
    }
    __syncthreads();

    // ---- write outputs (all POST_NMS slots, zero-filled when invalid) ----
    for (int s2 = tid; s2 < POST_NMS; s2 += NMS_THREADS) {
        int k = keepArr[s2];
        float px1 = 0.f, py1 = 0.f, px2 = 0.f, py2 = 0.f;
        float lv = 1e-08f, d0 = 0.f, d1 = 0.f, d2 = 0.f, d3 = 0.f, vv = 0.f;
        if (k >= 0) {
            unsigned oi = sortedIdx[k];
            float4 a = A[oi], d = D[oi];
            bool v;
            decode_box(a, d, W, H, px1, py1, px2, py2, v);
            lv = L[oi];
            d0 = d.x; d1 = d.y; d2 = d.z; d3 = d.w;
            vv = 1.0f;
        }
        size_t p4 = ((size_t)b * POST_NMS + s2) * 4;
        outP[p4 + 0] = px1; outP[p4 + 1] = py1; outP[p4 + 2] = px2; outP[p4 + 3] = py2;
        outL[(size_t)b * POST_NMS + s2] = lv;
        outD[p4 + 0] = d0; outD[p4 + 1] = d1; outD[p4 + 2] = d2; outD[p4 + 3] = d3;
        outV[(size_t)b * POST_NMS + s2] = vv;
    }
}

// ---------------- host launcher ----------------

extern "C" void kernel_launch(void* const* d_in, const int* in_sizes, int n_in,
                              void* d_out, int out_size, void* d_ws, size_t ws_size,
                              hipStream_t stream) {
    const float4* anch = (const float4*)d_in[0];   // (B,N,4) f32
    const int*    isz  = (const int*)d_in[1];      // (B,2) i32
    const float*  lgt  = (const float*)d_in[2];    // (B,N,1) f32
    const float4* dlt  = (const float4*)d_in[3];   // (B,N,4) f32

    const int B = in_sizes[1] / 2;
    const int N = in_sizes[2] / B;

    // workspace carve-out (~1.3 MB)
    int* hist = (int*)d_ws;                                // B*4096
    int* meta = hist + (size_t)B * NBUCKET;                // B
    int* cnt  = meta + B;                                  // B
    size_t packOfs = (((size_t)B * NBUCKET + 2 * (size_t)B) * sizeof(int) + 15) & ~(size_t)15;
    unsigned long long* candPacked = (unsigned long long*)((char*)d_ws + packOfs);  // B*CAND_CAP u64

    float* out  = (float*)d_out;
    float* outP = out;
    float* outL = outP + (size_t)B * POST_NMS * 4;
    float* outD = outL + (size_t)B * POST_NMS;
    float* outV = outD + (size_t)B * POST_NMS * 4;

    const int nZero = B * NBUCKET + 2 * B;  // hist + meta + cnt (contiguous)
    k_zero<<<(nZero + 255) / 256, 256, 0, stream>>>(hist, nZero);

    dim3 gridS(40, B);
    k_hist<<<gridS, 256, 0, stream>>>(anch, isz, lgt, dlt, N, hist);
    k_thresh<<<1, 32, 0, stream>>>(hist, meta, B);
    k_compact<<<gridS, 256, 0, stream>>>(anch, isz, lgt, dlt, N, meta, cnt, candPacked);
    k_sort_nms_out<<<B, NMS_THREADS, 0, stream>>>(anch, isz, lgt, dlt, N, cnt,
                                                  candPacked, outP, outL, outD, outV);
}